// S6_70480413327620
// MI455X (gfx1250) — compile-verified
//
#include <hip/hip_runtime.h>
#include <math.h>

// Problem constants (match reference)
constexpr int B_  = 2;
constexpr int L_  = 1024;
constexpr int DI  = 2048;   // d_inner
constexpr int NS  = 16;     // d_state
constexpr int R_  = 64;     // dt_rank
constexpr int PROJ = R_ + 2 * NS;   // 96
constexpr int M_  = B_ * L_;        // 2048 rows in the GEMMs
constexpr int CH  = 8;              // scan chunks
constexpr int LC  = L_ / CH;        // 128 timesteps per chunk

typedef __attribute__((ext_vector_type(2))) float v2f;
typedef __attribute__((ext_vector_type(4))) float v4f;
typedef __attribute__((ext_vector_type(8))) float v8f;

// ---------------------------------------------------------------------------
// Kernel 1: x_proj[M,96] = x[M,2048] @ W_x^T   (W_x is [96,2048] row-major)
// One wave per 16x16 output tile, fp32 WMMA, K stepped by 4.
// A layout (16x4 f32): lane<16 -> K={k,k+1}; lane>=16 -> K={k+2,k+3}; row=lane&15
// B layout (4x16 f32): mirrored; col=lane&15
// C/D layout: vgpr v, lane<16 -> M=v ; lane>=16 -> M=v+8 ; N=lane&15
// ---------------------------------------------------------------------------
__global__ __launch_bounds__(256) void s6_gemm_xproj(const float* __restrict__ x,
                                                     const float* __restrict__ Wx,
                                                     float* __restrict__ xp) {
    const int lane   = threadIdx.x & 31;
    const int waveId = threadIdx.x >> 5;
    const int tile   = blockIdx.x * 8 + waveId;      // 768 tiles total
    const int TN     = PROJ / 16;                    // 6
    const int tileM  = tile / TN;
    const int tileN  = tile % TN;

    const int row   = tileM * 16 + (lane & 15);
    const int col   = tileN * 16 + (lane & 15);
    const int khalf = (lane >> 4) * 2;

    v8f acc = {};
    const float* arow = x  + (size_t)row * DI;
    const float* brow = Wx + (size_t)col * DI;
#pragma unroll 4
    for (int k = 0; k < DI; k += 4) {
        const int kk = k + khalf;
        v2f a = *(const v2f*)(arow + kk);
        v2f b = *(const v2f*)(brow + kk);
        acc = __builtin_amdgcn_wmma_f32_16x16x4_f32(false, a, false, b,
                                                    (short)0, acc, false, false);
    }

    const int rbase = tileM * 16 + ((lane >> 4) * 8);
    const int cl    = tileN * 16 + (lane & 15);
#pragma unroll
    for (int v = 0; v < 8; ++v)
        xp[(size_t)(rbase + v) * PROJ + cl] = acc[v];
}

// ---------------------------------------------------------------------------
// Kernel 2: dt[M,2048] = softplus(x_proj[:, :64] @ W_dt^T + b_dt)
// W_dt is [2048,64] row-major. K=64.
// ---------------------------------------------------------------------------
__global__ __launch_bounds__(256) void s6_gemm_dt(const float* __restrict__ xp,
                                                  const float* __restrict__ Wdt,
                                                  const float* __restrict__ bdt,
                                                  float* __restrict__ dtm) {
    const int lane   = threadIdx.x & 31;
    const int waveId = threadIdx.x >> 5;
    const int tile   = blockIdx.x * 8 + waveId;      // 16384 tiles
    const int TN     = DI / 16;                      // 128
    const int tileM  = tile / TN;
    const int tileN  = tile % TN;

    const int row   = tileM * 16 + (lane & 15);
    const int col   = tileN * 16 + (lane & 15);
    const int khalf = (lane >> 4) * 2;

    v8f acc = {};
    const float* arow = xp  + (size_t)row * PROJ;    // dt_x = first 64 cols
    const float* brow = Wdt + (size_t)col * R_;
#pragma unroll
    for (int k = 0; k < R_; k += 4) {
        const int kk = k + khalf;
        v2f a = *(const v2f*)(arow + kk);
        v2f b = *(const v2f*)(brow + kk);
        acc = __builtin_amdgcn_wmma_f32_16x16x4_f32(false, a, false, b,
                                                    (short)0, acc, false, false);
    }

    const int rbase = tileM * 16 + ((lane >> 4) * 8);
    const int cl    = tileN * 16 + (lane & 15);
    const float bias = bdt[cl];
#pragma unroll
    for (int v = 0; v < 8; ++v) {
        float t = acc[v] + bias;
        // numerically stable softplus
        float sp = fmaxf(t, 0.0f) + log1pf(__expf(-fabsf(t)));
        dtm[(size_t)(rbase + v) * DI + cl] = sp;
    }
}

// ---------------------------------------------------------------------------
// Pass A: per chunk, scan from h=0 recording end-state h and sum(dt).
// One thread per d-channel; 16 states in registers.
// grid = (DI/256, CH, B)
// ---------------------------------------------------------------------------
__global__ __launch_bounds__(256) void s6_scan_stats(const float* __restrict__ x,
                                                     const float* __restrict__ xp,
                                                     const float* __restrict__ dtm,
                                                     const float* __restrict__ Alog,
                                                     float* __restrict__ hend,
                                                     float* __restrict__ dtsums) {
    const int d = blockIdx.x * blockDim.x + threadIdx.x;
    const int c = blockIdx.y;
    const int b = blockIdx.z;

    float Arow[NS];
#pragma unroll
    for (int n = 0; n < NS; ++n) Arow[n] = -expf(Alog[(size_t)d * NS + n]);

    float h[NS];
#pragma unroll
    for (int n = 0; n < NS; ++n) h[n] = 0.0f;
    float dtsum = 0.0f;

    const int t0 = c * LC;
    for (int t = t0; t < t0 + LC; ++t) {
        const int row = b * L_ + t;
        const float dtv = dtm[(size_t)row * DI + d];
        const float xv  = x  [(size_t)row * DI + d];
        dtsum += dtv;
        const float dx = dtv * xv;

        float Bv[NS];
        const v4f* Bp = (const v4f*)(xp + (size_t)row * PROJ + R_);
#pragma unroll
        for (int i = 0; i < 4; ++i) {
            v4f q = Bp[i];
#pragma unroll
            for (int j = 0; j < 4; ++j) Bv[4 * i + j] = q[j];
        }
#pragma unroll
        for (int n = 0; n < NS; ++n) {
            float dA = __expf(dtv * Arow[n]);
            h[n] = fmaf(dA, h[n], dx * Bv[n]);
        }
    }

    const size_t base = ((size_t)(b * CH + c) * DI + d);
#pragma unroll
    for (int n = 0; n < NS; ++n) hend[base * NS + n] = h[n];
    dtsums[base] = dtsum;
}

// ---------------------------------------------------------------------------
// Pass B: chain the CH chunk states per (b,d,n):
//   hin[c] = exp(A * dtsum[c-1]) * hin[c-1] + hend[c-1],  hin[0]=0
// One thread per (b,d,n) = 65536 threads.
// ---------------------------------------------------------------------------
__global__ __launch_bounds__(256) void s6_combine(const float* __restrict__ Alog,
                                                  const float* __restrict__ dtsums,
                                                  const float* __restrict__ hend,
                                                  float* __restrict__ hin) {
    const int idx = blockIdx.x * blockDim.x + threadIdx.x;   // < B*DI*NS
    const int b = idx >> 15;            // DI*NS = 32768
    const int d = (idx >> 4) & (DI - 1);
    const int n = idx & (NS - 1);

    const float Aval = -expf(Alog[(size_t)d * NS + n]);
    float h = 0.0f;
    hin[((size_t)(b * CH + 0) * DI + d) * NS + n] = 0.0f;
    for (int c = 1; c < CH; ++c) {
        const size_t prev = (size_t)(b * CH + c - 1) * DI + d;
        const float P = __expf(Aval * dtsums[prev]);
        h = fmaf(P, h, hend[prev * NS + n]);
        hin[((size_t)(b * CH + c) * DI + d) * NS + n] = h;
    }
}

// ---------------------------------------------------------------------------
// Pass C: re-scan each chunk from its true entry state, emit y.
// grid = (DI/256, CH, B)
// ---------------------------------------------------------------------------
__global__ __launch_bounds__(256) void s6_scan_final(const float* __restrict__ x,
                                                     const float* __restrict__ xp,
                                                     const float* __restrict__ dtm,
                                                     const float* __restrict__ Alog,
                                                     const float* __restrict__ Dvec,
                                                     const float* __restrict__ hin,
                                                     float* __restrict__ y) {
    const int d = blockIdx.x * blockDim.x + threadIdx.x;
    const int c = blockIdx.y;
    const int b = blockIdx.z;

    float Arow[NS];
#pragma unroll
    for (int n = 0; n < NS; ++n) Arow[n] = -expf(Alog[(size_t)d * NS + n]);
    const float Dval = Dvec[d];

    float h[NS];
    const size_t base = ((size_t)(b * CH + c) * DI + d) * NS;
#pragma unroll
    for (int n = 0; n < NS; ++n) h[n] = hin[base + n];

    const int t0 = c * LC;
    for (int t = t0; t < t0 + LC; ++t) {
        const int row = b * L_ + t;
        const float dtv = dtm[(size_t)row * DI + d];
        const float xv  = x  [(size_t)row * DI + d];
        const float dx  = dtv * xv;

        float Bv[NS], Cv[NS];
        const v4f* Bp = (const v4f*)(xp + (size_t)row * PROJ + R_);
        const v4f* Cp = (const v4f*)(xp + (size_t)row * PROJ + R_ + NS);
#pragma unroll
        for (int i = 0; i < 4; ++i) {
            v4f qb = Bp[i], qc = Cp[i];
#pragma unroll
            for (int j = 0; j < 4; ++j) { Bv[4 * i + j] = qb[j]; Cv[4 * i + j] = qc[j]; }
        }

        float yacc = 0.0f;
#pragma unroll
        for (int n = 0; n < NS; ++n) {
            float dA = __expf(dtv * Arow[n]);
            h[n] = fmaf(dA, h[n], dx * Bv[n]);
            yacc = fmaf(h[n], Cv[n], yacc);
        }
        y[(size_t)row * DI + d] = fmaf(xv, Dval, yacc);
    }
}

// ---------------------------------------------------------------------------
extern "C" void kernel_launch(void* const* d_in, const int* in_sizes, int n_in,
                              void* d_out, int out_size, void* d_ws, size_t ws_size,
                              hipStream_t stream) {
    (void)in_sizes; (void)n_in; (void)out_size; (void)ws_size;
    const float* x    = (const float*)d_in[0];   // [B,L,DI]
    const float* Wx   = (const float*)d_in[1];   // [96, DI]
    const float* Wdt  = (const float*)d_in[2];   // [DI, 64]
    const float* bdt  = (const float*)d_in[3];   // [DI]
    const float* Alog = (const float*)d_in[4];   // [DI, NS]
    const float* Dv   = (const float*)d_in[5];   // [DI]
    float* y = (float*)d_out;

    float* ws     = (float*)d_ws;
    float* xp     = ws;                                   // M*96
    float* dtm    = xp  + (size_t)M_ * PROJ;              // M*DI
    float* hend   = dtm + (size_t)M_ * DI;                // B*CH*DI*NS
    float* dtsums = hend + (size_t)B_ * CH * DI * NS;     // B*CH*DI
    float* hin    = dtsums + (size_t)B_ * CH * DI;        // B*CH*DI*NS

    // 1) x_proj GEMM: 768 tiles, 8 waves/block -> 96 blocks
    s6_gemm_xproj<<<(M_ / 16) * (PROJ / 16) / 8, 256, 0, stream>>>(x, Wx, xp);
    // 2) dt GEMM + softplus: 16384 tiles -> 2048 blocks
    s6_gemm_dt<<<(M_ / 16) * (DI / 16) / 8, 256, 0, stream>>>(xp, Wdt, bdt, dtm);
    // 3) chunk-local scans (stats)
    s6_scan_stats<<<dim3(DI / 256, CH, B_), 256, 0, stream>>>(x, xp, dtm, Alog, hend, dtsums);
    // 4) chain chunk states
    s6_combine<<<(B_ * DI * NS) / 256, 256, 0, stream>>>(Alog, dtsums, hend, hin);
    // 5) final scan + output
    s6_scan_final<<<dim3(DI / 256, CH, B_), 256, 0, stream>>>(x, xp, dtm, Alog, Dv, hin, y);
}